// WideDeepSimpleRNN_74062416053059
// MI455X (gfx1250) — compile-verified
//
#include <hip/hip_runtime.h>

#define H 128
#define INW 2048
#define PROJ 64
#define BSZ 64
#define TLEN 512
#define M_TOT (BSZ * TLEN)   // 32768
#define EPSV 1e-5f

typedef __attribute__((ext_vector_type(2))) float v2f;
typedef __attribute__((ext_vector_type(8))) float v8f;

// fp32 WMMA: D(16x16) = A(16x4) * B(4x16) + C ; 8-arg form per probe
#define WMMA_F32(a, b, c) \
  __builtin_amdgcn_wmma_f32_16x16x4_f32(false, (a), false, (b), (short)0, (c), false, false)

// ---------------------------------------------------------------------------
// Kernel 1: xp1[m, n] = x[m, :] . W_ih1[n, :] + (b_ih1[n] + b_hh1[n])
//   M = B*T = 32768, N = H = 128, K = IN = 2048
//   WG = 256 threads (8 waves). Each WG: 32-row strip, all 128 cols.
//   K staged through LDS in chunks of 128 with +4-float row padding.
// ---------------------------------------------------------------------------
#define KC 128
#define LW (KC + 4)   // 132: bank pattern 4L+k -> conflict free for frag reads

__global__ __launch_bounds__(256) void gemm1_kernel(
    const float* __restrict__ x, const float* __restrict__ Wih1,
    const float* __restrict__ bih1, const float* __restrict__ bhh1,
    float* __restrict__ xp1) {
  __shared__ float xs[32 * LW];
  __shared__ float ws[128 * LW];
  const int tid  = threadIdx.x;
  const int wave = tid >> 5;
  const int lane = tid & 31;
  const int half = lane >> 4;
  const int l16  = lane & 15;
  const int n0   = wave * 16;
  const long m0  = (long)blockIdx.x * 32;

  v8f c0 = {};
  v8f c1 = {};
  for (int k0 = 0; k0 < INW; k0 += KC) {
    // cooperative coalesced loads: x chunk 32xKC, W chunk 128xKC (float4)
    for (int i = 0; i < 4; ++i) {
      int idx = tid + i * 256;          // 0..1023
      int row = idx >> 5;
      int cv  = (idx & 31) << 2;
      float4 v = *(const float4*)&x[(m0 + row) * INW + k0 + cv];
      *(float4*)&xs[row * LW + cv] = v;
    }
    for (int i = 0; i < 16; ++i) {
      int idx = tid + i * 256;          // 0..4095
      int row = idx >> 5;
      int cv  = (idx & 31) << 2;
      float4 v = *(const float4*)&Wih1[(long)row * INW + k0 + cv];
      *(float4*)&ws[row * LW + cv] = v;
    }
    __syncthreads();
    for (int k = 0; k < KC; k += 4) {
      const int kc = k + 2 * half;
      v2f b;  b.x  = ws[(n0 + l16) * LW + kc];  b.y  = ws[(n0 + l16) * LW + kc + 1];
      v2f a0; a0.x = xs[l16 * LW + kc];         a0.y = xs[l16 * LW + kc + 1];
      v2f a1; a1.x = xs[(16 + l16) * LW + kc];  a1.y = xs[(16 + l16) * LW + kc + 1];
      c0 = WMMA_F32(a0, b, c0);
      c1 = WMMA_F32(a1, b, c1);
    }
    __syncthreads();
  }
  const float bias = bih1[n0 + l16] + bhh1[n0 + l16];
  for (int v = 0; v < 8; ++v) {
    long m = m0 + v + 8 * half;
    xp1[m * H + n0 + l16]        = c0[v] + bias;
    xp1[(m + 16) * H + n0 + l16] = c1[v] + bias;
  }
}

// ---------------------------------------------------------------------------
// Kernel 2: layer-1 recurrence fused with layer-2 input GEMM.
//   4 WGs x 256 threads; WG g owns batch rows [16g, 16g+16).
//   Per step t: h = tanh(xp1[:,t] + h @ Whh1^T); xp2[:,t] = h @ Wih2^T + b2
//   Whh1 / Wih2 B-fragments live in registers; h lives in LDS (padded rows).
// ---------------------------------------------------------------------------
#define HL (H + 4)   // 132

__global__ __launch_bounds__(256) void rnn1_kernel(
    const float* __restrict__ xp1, const float* __restrict__ Whh1,
    const float* __restrict__ Wih2, const float* __restrict__ bih2,
    const float* __restrict__ bhh2, float* __restrict__ xp2) {
  __shared__ float hs[16 * HL];
  const int tid  = threadIdx.x;
  const int wave = tid >> 5;
  const int lane = tid & 31;
  const int half = lane >> 4;
  const int l16  = lane & 15;
  const int n0   = wave * 16;
  const int b0   = blockIdx.x * 16;

  v2f bh[32], bi[32];
  for (int kk = 0; kk < 32; ++kk) {
    const int kc = kk * 4 + 2 * half;
    bh[kk].x = Whh1[(n0 + l16) * H + kc];  bh[kk].y = Whh1[(n0 + l16) * H + kc + 1];
    bi[kk].x = Wih2[(n0 + l16) * H + kc];  bi[kk].y = Wih2[(n0 + l16) * H + kc + 1];
  }
  const float bias2 = bih2[n0 + l16] + bhh2[n0 + l16];

  for (int i = tid; i < 16 * HL; i += 256) hs[i] = 0.0f;   // h0 = 0
  __syncthreads();

  for (int t = 0; t < TLEN; ++t) {
    v8f c;
    for (int v = 0; v < 8; ++v) {
      long m = (long)(b0 + v + 8 * half) * TLEN + t;
      c[v] = xp1[m * H + n0 + l16];
    }
    for (int kk = 0; kk < 32; ++kk) {
      const int kc = kk * 4 + 2 * half;
      v2f a; a.x = hs[l16 * HL + kc]; a.y = hs[l16 * HL + kc + 1];
      c = WMMA_F32(a, bh[kk], c);
    }
    for (int v = 0; v < 8; ++v) c[v] = tanhf(c[v]);
    __syncthreads();   // all waves done reading old h
    for (int v = 0; v < 8; ++v) hs[(v + 8 * half) * HL + n0 + l16] = c[v];
    __syncthreads();   // new h visible

    v8f c2;
    for (int v = 0; v < 8; ++v) c2[v] = bias2;
    for (int kk = 0; kk < 32; ++kk) {
      const int kc = kk * 4 + 2 * half;
      v2f a; a.x = hs[l16 * HL + kc]; a.y = hs[l16 * HL + kc + 1];
      c2 = WMMA_F32(a, bi[kk], c2);
    }
    for (int v = 0; v < 8; ++v) {
      long m = (long)(b0 + v + 8 * half) * TLEN + t;
      xp2[m * H + n0 + l16] = c2[v];
    }
  }
}

// ---------------------------------------------------------------------------
// Kernel 3: layer-2 recurrence, keep only final hidden state h2[64,128].
// ---------------------------------------------------------------------------
__global__ __launch_bounds__(256) void rnn2_kernel(
    const float* __restrict__ xp2, const float* __restrict__ Whh2,
    float* __restrict__ h2) {
  __shared__ float hs[16 * HL];
  const int tid  = threadIdx.x;
  const int wave = tid >> 5;
  const int lane = tid & 31;
  const int half = lane >> 4;
  const int l16  = lane & 15;
  const int n0   = wave * 16;
  const int b0   = blockIdx.x * 16;

  v2f bh[32];
  for (int kk = 0; kk < 32; ++kk) {
    const int kc = kk * 4 + 2 * half;
    bh[kk].x = Whh2[(n0 + l16) * H + kc];  bh[kk].y = Whh2[(n0 + l16) * H + kc + 1];
  }
  for (int i = tid; i < 16 * HL; i += 256) hs[i] = 0.0f;
  __syncthreads();

  v8f c = {};
  for (int t = 0; t < TLEN; ++t) {
    for (int v = 0; v < 8; ++v) {
      long m = (long)(b0 + v + 8 * half) * TLEN + t;
      c[v] = xp2[m * H + n0 + l16];
    }
    for (int kk = 0; kk < 32; ++kk) {
      const int kc = kk * 4 + 2 * half;
      v2f a; a.x = hs[l16 * HL + kc]; a.y = hs[l16 * HL + kc + 1];
      c = WMMA_F32(a, bh[kk], c);
    }
    for (int v = 0; v < 8; ++v) c[v] = tanhf(c[v]);
    __syncthreads();
    for (int v = 0; v < 8; ++v) hs[(v + 8 * half) * HL + n0 + l16] = c[v];
    __syncthreads();
  }
  for (int v = 0; v < 8; ++v)
    h2[(long)(b0 + v + 8 * half) * H + n0 + l16] = c[v];
}

// ---------------------------------------------------------------------------
// Kernel 4: z = h2 @ Wproj^T + bproj ; LayerNorm over last dim (64).
//   grid = 64 (one block per batch row), block = 64 threads (2 waves).
// ---------------------------------------------------------------------------
__global__ __launch_bounds__(64) void proj_ln_kernel(
    const float* __restrict__ h2, const float* __restrict__ Wp,
    const float* __restrict__ bp, const float* __restrict__ gamma,
    const float* __restrict__ beta, float* __restrict__ out) {
  __shared__ float tmp[PROJ];
  const int b = blockIdx.x;
  const int j = threadIdx.x;
  float acc = bp[j];
  for (int k = 0; k < H; ++k) acc += h2[b * H + k] * Wp[j * H + k];

  tmp[j] = acc;
  __syncthreads();
  for (int s = 32; s > 0; s >>= 1) {
    if (j < s) tmp[j] += tmp[j + s];
    __syncthreads();
  }
  const float mu = tmp[0] * (1.0f / PROJ);
  __syncthreads();
  const float d = acc - mu;
  tmp[j] = d * d;
  __syncthreads();
  for (int s = 32; s > 0; s >>= 1) {
    if (j < s) tmp[j] += tmp[j + s];
    __syncthreads();
  }
  const float var = tmp[0] * (1.0f / PROJ);
  const float r = rsqrtf(var + EPSV);
  out[b * PROJ + j] = d * r * gamma[j] + beta[j];
}

// ---------------------------------------------------------------------------
extern "C" void kernel_launch(void* const* d_in, const int* in_sizes, int n_in,
                              void* d_out, int out_size, void* d_ws, size_t ws_size,
                              hipStream_t stream) {
  (void)in_sizes; (void)n_in; (void)out_size; (void)ws_size;
  const float* x     = (const float*)d_in[0];
  const float* Wih1  = (const float*)d_in[1];
  const float* Whh1  = (const float*)d_in[2];
  const float* bih1  = (const float*)d_in[3];
  const float* bhh1  = (const float*)d_in[4];
  const float* Wih2  = (const float*)d_in[5];
  const float* Whh2  = (const float*)d_in[6];
  const float* bih2  = (const float*)d_in[7];
  const float* bhh2  = (const float*)d_in[8];
  const float* Wproj = (const float*)d_in[9];
  const float* bproj = (const float*)d_in[10];
  const float* gam   = (const float*)d_in[11];
  const float* bet   = (const float*)d_in[12];
  float* out = (float*)d_out;

  float* xp1 = (float*)d_ws;                       // 32768*128 f32 = 16 MB
  float* xp2 = xp1 + (size_t)M_TOT * H;            // 16 MB
  float* h2  = xp2 + (size_t)M_TOT * H;            // 32 KB

  gemm1_kernel<<<M_TOT / 32, 256, 0, stream>>>(x, Wih1, bih1, bhh1, xp1);
  rnn1_kernel<<<BSZ / 16, 256, 0, stream>>>(xp1, Whh1, Wih2, bih2, bhh2, xp2);
  rnn2_kernel<<<BSZ / 16, 256, 0, stream>>>(xp2, Whh2, h2);
  proj_ln_kernel<<<BSZ, PROJ, 0, stream>>>(h2, Wproj, bproj, gam, bet, out);
}